// EIRL_18880676233906
// MI455X (gfx1250) — compile-verified
//
#include <hip/hip_runtime.h>
#include <hip/hip_bf16.h>

typedef __attribute__((ext_vector_type(2))) float vf2;
typedef __attribute__((ext_vector_type(8))) float v8f;

#define NPN   50000            // = NP_ = NL_
#define DIM   128              // = DP = DL = OUT
#define NEXP  4
#define NEDGE 1600000
#define KDIM  256              // DP + DL
#define PER   (NEXP * NPN * DIM)   // 25,600,000 floats per (side) msg/y block

// workspace layout (float offsets). Total = 4*PER + 4096 floats = ~409.6 MB
#define OFF_MSG_L 0
#define OFF_MSG_P (PER)
#define OFF_SUM   (2*PER)
#define OFF_SUMSQ (2*PER + 1024)
#define OFF_MEAN  (2*PER + 2048)
#define OFF_INV   (2*PER + 3072)
#define OFF_Y_L   (2*PER + 4096)
#define OFF_Y_P   (OFF_Y_L + PER)
#define ZERO_F4   ((2*PER + 2048) / 4)   // msg_l + msg_p + sum + sumsq

// ---------------------------------------------------------------- zero scratch
__global__ void EIRL_zero(float4* __restrict__ p, long n4) {
  long i = (long)blockIdx.x * blockDim.x + threadIdx.x;
  long stride = (long)gridDim.x * blockDim.x;
  float4 z = make_float4(0.f, 0.f, 0.f, 0.f);
  for (; i < n4; i += stride) p[i] = z;
}

// ------------------------------------------------- fused 4-expert edge scatter
// 2 edges per 256-thread block; 128 threads cover the 128-dim feature vector.
// Reads px[row]/lx[col] once, scatters v[i]*feat into 4 expert msg buffers
// in both directions via global f32 atomics (L2-resident tables).
__global__ __launch_bounds__(256) void EIRL_edge_scatter(
    const float* __restrict__ px, const float* __restrict__ lx,
    const float* __restrict__ vals, const int* __restrict__ rows,
    const int* __restrict__ cols, float* __restrict__ msg_l,
    float* __restrict__ msg_p) {
  int e = blockIdx.x * 2 + (threadIdx.x >> 7);
  int t = threadIdx.x & 127;
  if (e >= NEDGE) return;
  int r = rows[e];
  int c = cols[e];
  float pxv = px[(size_t)r * DIM + t];
  float lxv = lx[(size_t)c * DIM + t];
#pragma unroll
  for (int i = 0; i < NEXP; ++i) {
    float v = vals[(size_t)i * NEDGE + e];
    atomicAdd(&msg_l[((size_t)i * NPN + c) * DIM + t], v * pxv);  // pl_mat^T @ px
    atomicAdd(&msg_p[((size_t)i * NPN + r) * DIM + t], v * lxv);  // pl_mat   @ lx
  }
}

// ---------------------------------- WMMA GEMM + bias + fused BN stats epilogue
// grid = (3125 row tiles, 8 = side*4 + expert). Block = 256 (8 wave32).
// Each block: stage X tile (16 rows x 256 cols, cat(msg,node)) into LDS
// (stride 260 -> conflict-free A reads), each wave owns a 16-wide output
// column tile and runs 64x V_WMMA_F32_16X16X4_F32.
__global__ __launch_bounds__(256) void EIRL_gemm_bn_stats(
    const float* __restrict__ px, const float* __restrict__ lx,
    const float* __restrict__ W_lp, const float* __restrict__ b_lp,
    const float* __restrict__ W_pl, const float* __restrict__ b_pl,
    float* __restrict__ ws) {
  const int s = blockIdx.y >> 2;   // 0 = ligand side (x_l), 1 = protein side (x_p)
  const int i = blockIdx.y & 3;    // expert
  const float* msg  = ws + (s ? OFF_MSG_P : OFF_MSG_L) + (size_t)i * NPN * DIM;
  const float* node = s ? px : lx;
  const float* W    = (s ? W_pl : W_lp) + (size_t)i * DIM * KDIM;
  const float* bias = (s ? b_pl : b_lp) + i * DIM;
  float* y    = ws + (s ? OFF_Y_P : OFF_Y_L) + (size_t)i * NPN * DIM;
  float* csum = ws + OFF_SUM   + (s * NEXP + i) * DIM;
  float* csq  = ws + OFF_SUMSQ + (s * NEXP + i) * DIM;

  __shared__ __align__(16) float Xs[16 * 260];
  const int nBase = blockIdx.x * 16;  // 50000 % 16 == 0 -> no guards, EXEC all 1s

  {  // stage X tile: X[r][k] = k<128 ? msg[n][k] : node[n][k-128]
    int r = threadIdx.x >> 4, cth = threadIdx.x & 15;
    int nr = nBase + r;
#pragma unroll
    for (int q = 0; q < 4; ++q) {
      int k = (cth + 16 * q) * 4;
      float4 v = (k < DIM)
                     ? *(const float4*)&msg[(size_t)nr * DIM + k]
                     : *(const float4*)&node[(size_t)nr * DIM + (k - DIM)];
      *(float4*)&Xs[r * 260 + k] = v;
    }
  }
  __syncthreads();

  const int lane = threadIdx.x & 31;
  const int wv   = threadIdx.x >> 5;       // wave id 0..7 -> output col tile
  const int m    = lane & 15;              // A row / B column within tile
  const int kh   = (lane >> 4) << 1;       // K pair select: low lanes K0/K1, high K2/K3
  const int o    = wv * 16 + m;            // global output column 0..127
  v8f c = {};

#if defined(__gfx1250__) && __has_builtin(__builtin_amdgcn_wmma_f32_16x16x4_f32)
#pragma unroll 8
  for (int k0 = 0; k0 < KDIM; k0 += 4) {
    vf2 a = *(const vf2*)&Xs[m * 260 + k0 + kh];              // ds_load_b64
    vf2 b = *(const vf2*)&W[(size_t)o * KDIM + k0 + kh];      // global_load_b64
    c = __builtin_amdgcn_wmma_f32_16x16x4_f32(
        /*neg_a=*/false, a, /*neg_b=*/false, b,
        /*c_mod=*/(short)0, c, /*reuse_a=*/false, /*reuse_b=*/false);
  }
#else
  {  // scalar fallback matching the WMMA C/D lane layout
    const int mOffF = (lane >> 4) * 8;
#pragma unroll
    for (int r = 0; r < 8; ++r) {
      float acc = 0.f;
      for (int k = 0; k < KDIM; ++k)
        acc += Xs[(mOffF + r) * 260 + k] * W[(size_t)o * KDIM + k];
      c[r] = acc;
    }
  }
#endif

  // epilogue: bias, store y, fused per-column sums for BatchNorm stats
  const int mOff = (lane >> 4) * 8;  // lanes 16-31 hold rows M=8..15
  float bo = bias[o];
  float ps = 0.f, psq = 0.f;
#pragma unroll
  for (int r = 0; r < 8; ++r) {
    float val = c[r] + bo;
    y[(size_t)(nBase + mOff + r) * DIM + o] = val;
    ps += val;
    psq += val * val;
  }
  // lanes l and l^16 hold the same column o -> combine to full 16-row colsum
  ps  += __shfl_xor(ps, 16);
  psq += __shfl_xor(psq, 16);
  if (lane < 16) {
    atomicAdd(&csum[o], ps);
    atomicAdd(&csq[o], psq);
  }
}

// ------------------------------------------------------------- BN finalize
__global__ void EIRL_bn_finalize(float* __restrict__ ws) {
  int idx = blockIdx.x * blockDim.x + threadIdx.x;
  if (idx >= 2 * NEXP * DIM) return;
  float n  = (float)NPN;
  float mu = ws[OFF_SUM + idx] / n;
  float var = ws[OFF_SUMSQ + idx] / n - mu * mu;  // biased variance (training BN)
  ws[OFF_MEAN + idx] = mu;
  ws[OFF_INV + idx]  = rsqrtf(var + 1e-5f);
}

// --------------------------- BN + ReLU + expert-weighted combine (streaming)
__global__ __launch_bounds__(256) void EIRL_combine(
    const float* __restrict__ y, const float* __restrict__ g,
    const float* __restrict__ be, const float* __restrict__ mean,
    const float* __restrict__ inv, const float* __restrict__ wvec,
    const float* __restrict__ bscal, float* __restrict__ out) {
  size_t idx = (size_t)blockIdx.x * blockDim.x + threadIdx.x;  // one float4
  if (idx >= (size_t)NPN * DIM / 4) return;
  int n = (int)(idx >> 5);
  int o = (int)(idx & 31) * 4;
  float b0 = bscal[0];
  float acc[4] = {b0, b0, b0, b0};
#pragma unroll
  for (int i = 0; i < NEXP; ++i) {
    float wi = wvec[i];
    float4 yv = *(const float4*)&y[((size_t)i * NPN + n) * DIM + o];
    float yy[4] = {yv.x, yv.y, yv.z, yv.w};
#pragma unroll
    for (int cc = 0; cc < 4; ++cc) {
      int oc = i * DIM + o + cc;
      float t = g[oc] * (yy[cc] - mean[oc]) * inv[oc] + be[oc];
      acc[cc] += wi * fmaxf(t, 0.f);
    }
  }
  *(float4*)&out[(size_t)n * DIM + o] =
      make_float4(acc[0], acc[1], acc[2], acc[3]);
}

extern "C" void kernel_launch(void* const* d_in, const int* in_sizes, int n_in,
                              void* d_out, int out_size, void* d_ws,
                              size_t ws_size, hipStream_t stream) {
  (void)in_sizes; (void)n_in; (void)out_size; (void)ws_size;
  const float* px    = (const float*)d_in[0];
  const float* lx    = (const float*)d_in[1];
  const float* vals  = (const float*)d_in[2];
  const float* W_lp  = (const float*)d_in[3];
  const float* b_lp  = (const float*)d_in[4];
  const float* g_lp  = (const float*)d_in[5];
  const float* be_lp = (const float*)d_in[6];
  const float* W_pl  = (const float*)d_in[7];
  const float* b_pl  = (const float*)d_in[8];
  const float* g_pl  = (const float*)d_in[9];
  const float* be_pl = (const float*)d_in[10];
  const float* wp    = (const float*)d_in[11];
  const float* bp    = (const float*)d_in[12];
  const float* wl    = (const float*)d_in[13];
  const float* bl    = (const float*)d_in[14];
  const int*   rows  = (const int*)d_in[15];
  const int*   cols  = (const int*)d_in[16];
  float* ws  = (float*)d_ws;
  float* out = (float*)d_out;

  // 1) zero msg buffers + stats accumulators (must happen every call)
  EIRL_zero<<<2048, 256, 0, stream>>>((float4*)ws, (long)ZERO_F4);

  // 2) fused 4-expert, 2-direction edge scatter
  EIRL_edge_scatter<<<NEDGE / 2, 256, 0, stream>>>(
      px, lx, vals, rows, cols, ws + OFF_MSG_L, ws + OFF_MSG_P);

  // 3) WMMA GEMM + bias + BN stats, all 8 (side, expert) problems
  dim3 gg(NPN / 16, 8);
  EIRL_gemm_bn_stats<<<gg, 256, 0, stream>>>(px, lx, W_lp, b_lp, W_pl, b_pl, ws);

  // 4) finalize mean / rsqrt(var+eps)
  EIRL_bn_finalize<<<(2 * NEXP * DIM + 255) / 256, 256, 0, stream>>>(ws);

  // 5) BN + ReLU + weighted expert combine
  const int cblocks = (NPN * DIM / 4 + 255) / 256;
  // px_rep <- protein side (s=1): y_p, g_pl/be_pl, wp, bp
  EIRL_combine<<<cblocks, 256, 0, stream>>>(
      ws + OFF_Y_P, g_pl, be_pl, ws + OFF_MEAN + 1 * NEXP * DIM,
      ws + OFF_INV + 1 * NEXP * DIM, wp, bp, out);
  // lx_rep <- ligand side (s=0): y_l, g_lp/be_lp, wl, bl
  EIRL_combine<<<cblocks, 256, 0, stream>>>(
      ws + OFF_Y_L, g_lp, be_lp, ws + OFF_MEAN + 0 * NEXP * DIM,
      ws + OFF_INV + 0 * NEXP * DIM, wl, bl, out + (size_t)NPN * DIM);
}